// LWR_batch_version_5334349382155
// MI455X (gfx1250) — compile-verified
//
#include <hip/hip_runtime.h>
#include <stdint.h>

#define LWR_NX      1024
#define LWR_THREADS 128
#define LWR_CPT     8                 // cells per thread: 128*8 = 1024
#define LWR_TSKIP   8
#define LWR_NTMAX   1024
#define LWR_CFL     ((float)(0.0016 * 1.0 / 0.1))

// gfx1250 async global->LDS. Probe result: param 0 is v4i in AS1 (printed as
// "__device__" in HIP diagnostics), param 1 is the LDS-side pointer (AS3).
#if defined(__AMDGCN__) && __has_builtin(__builtin_amdgcn_global_load_async_to_lds_b128)
#define LWR_ASYNC 1
#else
#define LWR_ASYNC 0
#endif

#if LWR_ASYNC
typedef int lwr_v4i __attribute__((vector_size(16)));
typedef __attribute__((address_space(1))) lwr_v4i lwr_gv4i;
typedef __attribute__((address_space(3))) lwr_v4i lwr_lv4i;

__device__ __forceinline__ void lwr_async_b128(const void* g, void* l) {
  __builtin_amdgcn_global_load_async_to_lds_b128(
      (lwr_gv4i*)g, (lwr_lv4i*)l, 0, 0);
}
#endif

extern "C" __global__ void __launch_bounds__(LWR_THREADS, 1)
lwr_scan_kernel(const float* __restrict__ init,   // (B,3,NX)
                const float* __restrict__ bnd,    // (B,3,NT)
                const float* __restrict__ kappa,  // (NX)
                const float* __restrict__ vf,     // (NX)
                const int*   __restrict__ sidx,   // (NS)
                const int*   __restrict__ tsp,    // scalar tsteps
                float* __restrict__ out,          // (B,3,nsave,NS-1)
                int NT, int NS)
{
  __shared__ __attribute__((aligned(16))) float s_bnd[3 * LWR_NTMAX];
  __shared__ float s_vf[LWR_NX];
  __shared__ float s_ik[LWR_NX];
  __shared__ float s_k [LWR_NX];
  __shared__ float s_qe[2][LWR_THREADS];   // double-buffered right-edge q

  const int tid = (int)threadIdx.x;
  const int b   = (int)blockIdx.x;
  const int c0  = tid * LWR_CPT;

  // ---- stage boundary rows for this batch into LDS (async TDM-style path) ----
  const float* brow = bnd + (size_t)b * 3u * (size_t)NT;
  const int nb = 3 * NT;                    // floats
#if LWR_ASYNC
  {
    const char* g = (const char*)brow;
    char*       l = (char*)s_bnd;
    const int nbytes16 = (nb * 4) & ~15;
    for (int off = tid * 16; off < nbytes16; off += LWR_THREADS * 16)
      lwr_async_b128(g + off, l + off);
    for (int t = (nbytes16 >> 2) + tid; t < nb; t += LWR_THREADS)
      s_bnd[t] = brow[t];
  }
#else
  for (int t = tid; t < nb; t += LWR_THREADS) s_bnd[t] = brow[t];
#endif

  // ---- per-thread state in registers ----
  float kk[LWR_CPT], qq[LWR_CPT], av[LWR_CPT], bv[LWR_CPT];
  const float* ik0 = init + (size_t)b * 3u * (size_t)LWR_NX;
  const float* iq0 = ik0 + LWR_NX;
  #pragma unroll
  for (int j = 0; j < LWR_CPT; ++j) {
    const int i = c0 + j;
    kk[j] = ik0[i];
    qq[j] = iq0[i];
    const float vfi = vf[i];
    const float ika = 1.0f / kappa[i];
    av[j] = vfi;
    bv[j] = vfi * ika;      // nu = av - bv*nk
    s_vf[i] = vfi;
    s_ik[i] = ika;
  }

  int my_si = 0;
  if (tid < NS - 1) my_si = sidx[tid + 1];

  const int steps = tsp[0] - 1;
  const int nsave = (steps >= LWR_TSKIP) ? ((steps - LWR_TSKIP) / LWR_TSKIP + 1) : 0;

#if LWR_ASYNC
  asm volatile("s_wait_asynccnt 0" ::: "memory");
#endif
  __syncthreads();

  // ---- serial time loop, state fully resident on-chip ----
  for (int s = 0; s < steps; ++s) {
    // publish right edge (double-buffered -> single barrier per step is safe)
    s_qe[s & 1][tid] = qq[LWR_CPT - 1];
#if defined(__AMDGCN__)
    // split barrier: overlap interior-cell math with barrier latency
    asm volatile("s_wait_dscnt 0\n\ts_barrier_signal -1" ::: "memory");
#else
    __syncthreads();
#endif
    float nk[LWR_CPT], nq[LWR_CPT];
    #pragma unroll
    for (int j = 1; j < LWR_CPT; ++j) {       // cells that need no neighbor
      const float v = fmaf(-LWR_CFL, qq[j] - qq[j - 1], kk[j]);
      const float u = fmaf(-bv[j], v, av[j]);
      nk[j] = v;
      nq[j] = v * u;
    }
#if defined(__AMDGCN__)
    asm volatile("s_barrier_wait -1" ::: "memory");
#endif
    if (tid == 0) {                            // global left boundary cell
      nk[0] = s_bnd[s + 1];
      nq[0] = s_bnd[NT + s + 1];
    } else {
      const float ql = s_qe[s & 1][tid - 1];
      const float v  = fmaf(-LWR_CFL, qq[0] - ql, kk[0]);
      const float u  = fmaf(-bv[0], v, av[0]);
      nk[0] = v;
      nq[0] = v * u;
    }
    #pragma unroll
    for (int j = 0; j < LWR_CPT; ++j) { kk[j] = nk[j]; qq[j] = nq[j]; }

    // ---- sensor sampling every TSKIP steps ----
    if ((s & (LWR_TSKIP - 1)) == (LWR_TSKIP - 1)) {
      #pragma unroll
      for (int j = 0; j < LWR_CPT; ++j) s_k[c0 + j] = kk[j];
      __syncthreads();
      if (tid < NS - 1) {
        const int i = my_si;
        float kv, qv, uv;
        if (i == 0) {
          kv = s_bnd[s + 1];
          qv = s_bnd[NT + s + 1];
          uv = s_bnd[2 * NT + s + 1];
        } else {
          kv = s_k[i];
          uv = s_vf[i] * (1.0f - kv * s_ik[i]);
          qv = kv * uv;
        }
        const int    ts     = (s - (LWR_TSKIP - 1)) >> 3;     // TSKIP == 8
        const size_t base   = (size_t)b * 3u * (size_t)nsave;
        const size_t stride = (size_t)(NS - 1);
        out[(base +                              (size_t)ts) * stride + tid] = kv; // ch 0: k
        out[(base +       (size_t)nsave        + (size_t)ts) * stride + tid] = qv; // ch 1: q
        out[(base + 2u * (size_t)nsave         + (size_t)ts) * stride + tid] = uv; // ch 2: u
      }
    }
  }
}

extern "C" void kernel_launch(void* const* d_in, const int* in_sizes, int n_in,
                              void* d_out, int out_size, void* d_ws, size_t ws_size,
                              hipStream_t stream) {
  const float* init = (const float*)d_in[0];
  const float* bnd  = (const float*)d_in[1];
  const float* kap  = (const float*)d_in[2];
  const float* vf   = (const float*)d_in[3];
  const int*   sidx = (const int*)d_in[4];
  const int*   tsp  = (const int*)d_in[5];
  (void)n_in; (void)d_ws; (void)ws_size; (void)out_size;

  const int NX    = in_sizes[2];                 // 1024
  const int batch = in_sizes[0] / (3 * NX);      // 256
  const int NT    = in_sizes[1] / (3 * batch);   // 1024
  const int NS    = in_sizes[4];                 // 64

  dim3 grid((unsigned)batch), block(LWR_THREADS);
  lwr_scan_kernel<<<grid, block, 0, stream>>>(init, bnd, kap, vf, sidx, tsp,
                                              (float*)d_out, NT, NS);
}